// GCN_52183852646433
// MI455X (gfx1250) — compile-verified
//
#include <hip/hip_runtime.h>

// Problem constants (from reference)
#define NFEAT   256
#define NHID    128
#define NCLASS  40
#define NNODES  50000
#define NEDGES  800000
#define BN_EPS  1e-5f

typedef float v2f __attribute__((ext_vector_type(2)));
typedef float v8f __attribute__((ext_vector_type(8)));

// ---------------------------------------------------------------------------
// GEMM0: Hlin[50000,128] = X[50000,256] @ W0[256,128]  (fp32 WMMA 16x16x4)
// One wave computes one 16x16 tile. 3125 m-tiles x 8 n-tiles = 25000 tiles.
// ---------------------------------------------------------------------------
__global__ __launch_bounds__(256) void gemm_x_w0(const float* __restrict__ X,
                                                 const float* __restrict__ W0,
                                                 float* __restrict__ H) {
    const int lane = threadIdx.x & 31;
    const int wave = threadIdx.x >> 5;
    const int tile = blockIdx.x * 8 + wave;     // grid sized exactly: 3125 blocks
    const int m0 = (tile >> 3) << 4;
    const int n0 = (tile & 7) << 4;
    const int r15  = lane & 15;                 // A row / B col within tile
    const int half = lane >> 4;                 // lane half selects K pair / M+8

    const float* Arow = X  + (size_t)(m0 + r15) * NFEAT;
    const float* Bcol = W0 + n0 + r15;

    v8f c = {0.f, 0.f, 0.f, 0.f, 0.f, 0.f, 0.f, 0.f};
#pragma unroll 4
    for (int k0 = 0; k0 < NFEAT; k0 += 4) {
        const int kb = k0 + (half << 1);
        v2f a = *(const v2f*)(Arow + kb);                 // A[m][kb], A[m][kb+1]
        v2f b;
        b.x = Bcol[(size_t)kb * NHID];                    // B[kb][n]
        b.y = Bcol[(size_t)(kb + 1) * NHID];              // B[kb+1][n]
        c = __builtin_amdgcn_wmma_f32_16x16x4_f32(false, a, false, b,
                                                  (short)0, c, false, false);
    }
    // C layout: lanes 0-15 -> M=r, lanes 16-31 -> M=r+8; N = lane&15
    float* Crow = H + (size_t)(m0 + half * 8) * NHID + n0 + r15;
#pragma unroll
    for (int r = 0; r < 8; ++r) Crow[(size_t)r * NHID] = c[r];
}

// ---------------------------------------------------------------------------
// GEMM1: T[50000,40] = Hagg[50000,128] @ W1[128,40]  (N padded to 48, guarded)
// ---------------------------------------------------------------------------
__global__ __launch_bounds__(256) void gemm_h_w1(const float* __restrict__ H,
                                                 const float* __restrict__ W1,
                                                 float* __restrict__ T) {
    const int lane = threadIdx.x & 31;
    const int wave = threadIdx.x >> 5;
    const int tile = blockIdx.x * 8 + wave;
    if (tile >= 3125 * 3) return;               // wave-uniform guard
    const int m0 = (tile / 3) << 4;
    const int n0 = (tile % 3) << 4;
    const int r15  = lane & 15;
    const int half = lane >> 4;
    const int col  = n0 + r15;
    const bool colok = (col < NCLASS);

    const float* Arow = H + (size_t)(m0 + r15) * NHID;

    v8f c = {0.f, 0.f, 0.f, 0.f, 0.f, 0.f, 0.f, 0.f};
#pragma unroll 4
    for (int k0 = 0; k0 < NHID; k0 += 4) {
        const int kb = k0 + (half << 1);
        v2f a = *(const v2f*)(Arow + kb);
        v2f b;
        b.x = colok ? W1[(size_t)kb * NCLASS + col] : 0.f;
        b.y = colok ? W1[(size_t)(kb + 1) * NCLASS + col] : 0.f;
        c = __builtin_amdgcn_wmma_f32_16x16x4_f32(false, a, false, b,
                                                  (short)0, c, false, false);
    }
    if (colok) {
        float* Crow = T + (size_t)(m0 + half * 8) * NCLASS + col;
#pragma unroll
        for (int r = 0; r < 8; ++r) Crow[(size_t)r * NCLASS] = c[r];
    }
}

// ---------------------------------------------------------------------------
// Init kernels: accumulators start at their bias values; BN stats at zero.
// (Re-run every launch -> deterministic across graph replays.)
// ---------------------------------------------------------------------------
__global__ void init_hagg(float* __restrict__ Hagg, const float* __restrict__ b0) {
    int idx = blockIdx.x * blockDim.x + threadIdx.x;
    if (idx < NNODES * NHID) Hagg[idx] = b0[idx & (NHID - 1)];
}
__global__ void init_out(float* __restrict__ Out, const float* __restrict__ b1) {
    int idx = blockIdx.x * blockDim.x + threadIdx.x;
    if (idx < NNODES * NCLASS) Out[idx] = b1[idx % NCLASS];
}
__global__ void init_stats(float* __restrict__ s) {
    int i = blockIdx.x * blockDim.x + threadIdx.x;
    if (i < 512) s[i] = 0.f;
}

// ---------------------------------------------------------------------------
// Edge scatter: Hagg[dst] += Hlin[src] (128 feats -> 32 float4 chunks/edge)
// ---------------------------------------------------------------------------
__global__ __launch_bounds__(256) void scatter0(const long long* __restrict__ ei,
                                                const float* __restrict__ Hlin,
                                                float* __restrict__ Hagg) {
    int idx = blockIdx.x * blockDim.x + threadIdx.x;   // e*32 + chunk
    int e = idx >> 5;
    if (e >= NEDGES) return;
    int chunk = (idx & 31) << 2;
    int s = (int)ei[e];
    int d = (int)ei[NEDGES + e];
    const float4 v = *(const float4*)(Hlin + (size_t)s * NHID + chunk);
    float* p = Hagg + (size_t)d * NHID + chunk;
    unsafeAtomicAdd(p + 0, v.x);
    unsafeAtomicAdd(p + 1, v.y);
    unsafeAtomicAdd(p + 2, v.z);
    unsafeAtomicAdd(p + 3, v.w);
}

// Out[dst] += T[src] (40 feats -> 10 float4 chunks/edge)
__global__ __launch_bounds__(256) void scatter1(const long long* __restrict__ ei,
                                                const float* __restrict__ T,
                                                float* __restrict__ Out) {
    int idx = blockIdx.x * blockDim.x + threadIdx.x;   // e*10 + chunk
    int e = idx / 10;
    if (e >= NEDGES) return;
    int chunk = (idx % 10) << 2;
    int s = (int)ei[e];
    int d = (int)ei[NEDGES + e];
    const float4 v = *(const float4*)(T + (size_t)s * NCLASS + chunk);
    float* p = Out + (size_t)d * NCLASS + chunk;
    unsafeAtomicAdd(p + 0, v.x);
    unsafeAtomicAdd(p + 1, v.y);
    unsafeAtomicAdd(p + 2, v.z);
    unsafeAtomicAdd(p + 3, v.w);
}

// ---------------------------------------------------------------------------
// BatchNorm: per-feature sum / sumsq -> scale/shift -> fused normalize+ReLU
// ---------------------------------------------------------------------------
#define BN_ROWS_PER_BLOCK 250
__global__ __launch_bounds__(256) void bn_stats(const float* __restrict__ Hagg,
                                                float* __restrict__ stats) {
    __shared__ float ssum[256];
    __shared__ float ssq[256];
    const int tid = threadIdx.x;
    const int f   = tid & 127;
    const int sub = tid >> 7;                   // 0 or 1: interleave rows
    float s = 0.f, q = 0.f;
    const int row0 = blockIdx.x * BN_ROWS_PER_BLOCK;
    const int row1 = min(row0 + BN_ROWS_PER_BLOCK, NNODES);
    for (int r = row0 + sub; r < row1; r += 2) {
        float v = Hagg[(size_t)r * NHID + f];
        s += v;
        q += v * v;
    }
    ssum[tid] = s; ssq[tid] = q;
    __syncthreads();
    if (tid < 128) {
        s = ssum[tid] + ssum[tid + 128];
        q = ssq[tid] + ssq[tid + 128];
        unsafeAtomicAdd(&stats[f], s);
        unsafeAtomicAdd(&stats[128 + f], q);
    }
}

__global__ void bn_finalize(float* __restrict__ stats,
                            const float* __restrict__ gamma,
                            const float* __restrict__ beta) {
    int f = threadIdx.x;
    if (f < NHID) {
        float inv_n = 1.0f / (float)NNODES;
        float mean  = stats[f] * inv_n;
        float var   = stats[128 + f] * inv_n - mean * mean;
        float sc    = gamma[f] * rsqrtf(var + BN_EPS);
        stats[256 + f] = sc;
        stats[384 + f] = beta[f] - mean * sc;
    }
}

__global__ void bn_apply_relu(float* __restrict__ Hagg,
                              const float* __restrict__ stats) {
    int idx = blockIdx.x * blockDim.x + threadIdx.x;
    if (idx < NNODES * NHID) {
        int f = idx & 127;
        float v = Hagg[idx] * stats[256 + f] + stats[384 + f];
        Hagg[idx] = v > 0.f ? v : 0.f;
    }
}

// ---------------------------------------------------------------------------
// Launch
// ---------------------------------------------------------------------------
extern "C" void kernel_launch(void* const* d_in, const int* in_sizes, int n_in,
                              void* d_out, int out_size, void* d_ws, size_t ws_size,
                              hipStream_t stream) {
    const float*     x      = (const float*)d_in[0];
    const long long* ei     = (const long long*)d_in[1];   // int64 [2, E]
    const float*     W0     = (const float*)d_in[2];
    const float*     b0     = (const float*)d_in[3];
    const float*     gamma0 = (const float*)d_in[4];
    const float*     beta0  = (const float*)d_in[5];
    const float*     W1     = (const float*)d_in[6];
    const float*     b1     = (const float*)d_in[7];
    float*           out    = (float*)d_out;

    // Workspace layout (bytes): Hlin 25.6MB | Hagg 25.6MB | stats 2KB
    // T (layer-2 activations, 8MB) aliases Hlin (dead by then).
    char*  ws    = (char*)d_ws;
    float* Hlin  = (float*)(ws);
    float* Hagg  = (float*)(ws + 25600000);
    float* stats = (float*)(ws + 51200000);
    float* T     = Hlin;

    // Init accumulators with biases, zero BN stats
    init_hagg <<<25000, 256, 0, stream>>>(Hagg, b0);        // 6.4M elems
    init_out  <<<7813,  256, 0, stream>>>(out, b1);         // 2.0M elems
    init_stats<<<2,     256, 0, stream>>>(stats);

    // Layer 0
    gemm_x_w0 <<<3125,   256, 0, stream>>>(x, W0, Hlin);    // 25000 tiles / 8 waves
    scatter0  <<<100000, 256, 0, stream>>>(ei, Hlin, Hagg); // 800k edges * 32 chunks

    // BatchNorm + ReLU
    bn_stats     <<<200,   256, 0, stream>>>(Hagg, stats);
    bn_finalize  <<<1,     128, 0, stream>>>(stats, gamma0, beta0);
    bn_apply_relu<<<25000, 256, 0, stream>>>(Hagg, stats);

    // Layer 1
    gemm_h_w1 <<<1172,  256, 0, stream>>>(Hagg, W1, T);     // 9375 tiles / 8 waves
    scatter1  <<<31250, 256, 0, stream>>>(ei, T, out);      // 800k edges * 10 chunks
}